// ODENet_33517924778483
// MI455X (gfx1250) — compile-verified
//
#include <hip/hip_runtime.h>

// ---------------------------------------------------------------------------
// ODENet LogSigProdLayer fused kernel for MI455X (gfx1250, wave32).
//
// Dominant cost: streaming W (8192x8192 f32 = 256 MB) once from HBM.
// 256 MB / 23.3 TB/s ~= 11 us floor -> design is bandwidth-bound:
//   - 128-bit non-temporal loads (read-once data, keep L2 for logx/partials)
//   - one column per lane-component: column sums accumulate in-register,
//     zero cross-lane traffic
//   - 95% of W is zero and logsigmoid(0)+ln2 == 0, so predicate away the
//     exp/log for zero entries (transcendental work overlaps the streaming)
// Phase 2 reduces the 64x8192 partial-sum matrix with V_WMMA_F32_16X16X4_F32
// using an all-ones A matrix (D[r][c] = sum_k B[k][c] + C), which is immune
// to any K-slot permutation in the B register layout.
// ---------------------------------------------------------------------------

#define NDIM 8192
#define CHUNKS 64                         // row chunks (K dim of phase-2 WMMA)
#define ROWS_PER_CHUNK (NDIM / CHUNKS)    // 128
#define THREADS 256
#define COLS_PER_THREAD 4
#define COLS_PER_BLOCK (THREADS * COLS_PER_THREAD)  // 1024
#define COLBLOCKS (NDIM / COLS_PER_BLOCK)           // 8

typedef __attribute__((ext_vector_type(2))) float v2f;
typedef __attribute__((ext_vector_type(4))) float v4f;
typedef __attribute__((ext_vector_type(8))) float v8f;

// logsigmoid(m) + ln2, numerically stable for any m:
//   = ln2 + min(m,0) - log1p(exp(-|m|))
__device__ __forceinline__ float lsig_p_ln2(float m) {
    float e = __expf(-fabsf(m));                  // in (0, 1]
    return 0.69314718055994531f + fminf(m, 0.0f) - __logf(1.0f + e);
}

// ---- phase 0: logx[j] = log(relu(y[j]) + 1e-3) ----------------------------
__global__ __launch_bounds__(256)
void logx_kernel(const float* __restrict__ y, float* __restrict__ logx) {
    int j = blockIdx.x * blockDim.x + threadIdx.x;
    if (j < NDIM) logx[j] = __logf(fmaxf(y[j], 0.0f) + 1e-3f);
}

// ---- phase 1: partial column sums over 128-row chunks ---------------------
__global__ __launch_bounds__(THREADS)
void colsum_partial_kernel(const float* __restrict__ W,
                           const float* __restrict__ logx,
                           float* __restrict__ partial) {
    const int col0 = blockIdx.x * COLS_PER_BLOCK + threadIdx.x * COLS_PER_THREAD;
    const int row0 = blockIdx.y * ROWS_PER_CHUNK;

    v4f acc = {};
#pragma unroll 4
    for (int r = 0; r < ROWS_PER_CHUNK; ++r) {
        const int i = row0 + r;
        const float li = logx[i];   // uniform across block -> scalar load
        // read-once streaming data: non-temporal 128-bit load
        const v4f w = __builtin_nontemporal_load(
            (const v4f*)(W + (size_t)i * NDIM + col0));
        // zero entries contribute exactly 0; skip the transcendentals
        if (w.x != 0.0f) acc.x += lsig_p_ln2(li * w.x);
        if (w.y != 0.0f) acc.y += lsig_p_ln2(li * w.y);
        if (w.z != 0.0f) acc.z += lsig_p_ln2(li * w.z);
        if (w.w != 0.0f) acc.w += lsig_p_ln2(li * w.w);
    }
    *(v4f*)(partial + (size_t)blockIdx.y * NDIM + col0) = acc;
}

// ---- phase 2: WMMA ones-reduction over CHUNKS + epilogue ------------------
// partial is a [CHUNKS x NDIM] matrix; column sums = ones(1,CHUNKS) @ partial.
// Each wave (32 lanes) owns one 16-column tile and accumulates K=CHUNKS in
// steps of 4 with V_WMMA_F32_16X16X4_F32 (A = all-ones 16x4).
__global__ __launch_bounds__(256)
void reduce_epilogue_kernel(const float* __restrict__ partial,
                            const float* __restrict__ bias,
                            const float* __restrict__ gm,
                            const float* __restrict__ y,
                            float* __restrict__ out) {
    const int wave = threadIdx.x >> 5;
    const int lane = threadIdx.x & 31;
    const int base = blockIdx.x * 128 + wave * 16;     // 8 waves * 16 cols

    const v2f a = {1.0f, 1.0f};   // A matrix: all ones (every slot, every lane)
    v8f c = {};                   // C/D accumulator

    const int colOff = base + (lane & 15);
    // distinct K row per (vgpr, lane-half) slot; with ones-A any bijective
    // assignment of the 4 K rows to slots yields the same column sum.
    const int kHi = (lane < 16) ? 0 : 2;

    for (int kk = 0; kk < CHUNKS; kk += 4) {
        v2f b;
        b.x = partial[(size_t)(kk + kHi + 0) * NDIM + colOff];
        b.y = partial[(size_t)(kk + kHi + 1) * NDIM + colOff];
        // D = A x B + C  (16x16x4 f32)
        c = __builtin_amdgcn_wmma_f32_16x16x4_f32(
                /*neg_a=*/false, a, /*neg_b=*/false, b,
                /*c_mod=*/(short)0, c, /*reuse_a=*/false, /*reuse_b=*/false);
    }

    // All D rows are identical (A is ones): VGPR0, lanes 0-15 hold the
    // column sums for columns base+0 .. base+15.
    if (lane < 16) {
        const int j = base + lane;
        const float s = c[0];
        const float prods = (s + bias[j]) * 1e-4f;
        out[j] = fmaxf(gm[j], 0.0f) * (__expf(prods) - y[j]);
    }
}

// ---------------------------------------------------------------------------
extern "C" void kernel_launch(void* const* d_in, const int* in_sizes, int n_in,
                              void* d_out, int out_size, void* d_ws, size_t ws_size,
                              hipStream_t stream) {
    // inputs: 0=t(1), 1=y(N), 2=weight(NxN), 3=bias(N), 4=gene_multipliers(N)
    const float* y    = (const float*)d_in[1];
    const float* W    = (const float*)d_in[2];
    const float* bias = (const float*)d_in[3];
    const float* gm   = (const float*)d_in[4];
    float* out = (float*)d_out;

    // ws layout: [logx: NDIM floats][partial: CHUNKS*NDIM floats] (~2.03 MB)
    float* logx    = (float*)d_ws;
    float* partial = logx + NDIM;

    logx_kernel<<<NDIM / 256, 256, 0, stream>>>(y, logx);

    dim3 grid1(COLBLOCKS, CHUNKS);
    colsum_partial_kernel<<<grid1, THREADS, 0, stream>>>(W, logx, partial);

    reduce_epilogue_kernel<<<NDIM / 128, 256, 0, stream>>>(partial, bias, gm, y, out);
}